// GATHead_35476429865591
// MI455X (gfx1250) — compile-verified
//
#include <hip/hip_runtime.h>

typedef __attribute__((ext_vector_type(2))) float v2f;
typedef __attribute__((ext_vector_type(8))) float v8f;
typedef __attribute__((ext_vector_type(2))) int   v2i;

#define GAT_D 128
#define TILE_STRIDE 68          // 64 cols + 4 floats pad: conflict-free, 16B-aligned rows
#define NEG_SLOPE 0.2f

// CDNA5 async global->LDS path (ASYNCcnt-tracked), if this toolchain exposes it.
#if defined(__AMDGCN__) && __has_builtin(__builtin_amdgcn_global_load_async_to_lds_b64) && \
    __has_builtin(__builtin_amdgcn_s_wait_asynccnt)
#define HAVE_ASYNC_LDS 1
#else
#define HAVE_ASYNC_LDS 0
#endif

// Copy 8 bytes global->LDS, per lane.
static __device__ __forceinline__ void row_copy8(const float* __restrict__ gsrc,
                                                 float* ldst)
{
#if HAVE_ASYNC_LDS
    // Builtin signature (from diagnostics): (v2i AS1*, v2i AS3*, imm offset, imm cpol)
    __builtin_amdgcn_global_load_async_to_lds_b64(
        (__attribute__((address_space(1))) v2i*)(gsrc),
        (__attribute__((address_space(3))) v2i*)(ldst),
        0, 0);
#else
    *reinterpret_cast<float2*>(ldst) = *reinterpret_cast<const float2*>(gsrc);
#endif
}

// ---------------------------------------------------------------------------
// Phase 1: s_src[n] = dot(h[n], W1), s_dst[n] = dot(h[n], W2)
// Skinny GEMM (N x 128) * (128 x 2) via V_WMMA_F32_16X16X4_F32.
// One wave owns 16 nodes; K=128 consumed as 2 halves of 64, each half staged
// in LDS (async-to-LDS when available), then 16 WMMA steps of K=4.
// ---------------------------------------------------------------------------
__global__ __launch_bounds__(256) void gat_scores_kernel(
    const float* __restrict__ h, const float* __restrict__ W,
    float* __restrict__ ssrc, float* __restrict__ sdst, int N)
{
    // [0..127]=W1, [128..255]=W2, [256..383]=zeros (B rows 2..15 read here,
    // so every lane issues an unconditional ds_load_b64 -> EXEC stays all-1s).
    __shared__ float Wext[3 * GAT_D];
    __shared__ float tile[8][16 * TILE_STRIDE];      // per-wave 16x64(+pad) h tile

    const int tid  = threadIdx.x;
    const int wave = tid >> 5;
    const int lane = tid & 31;

    if (tid < 2 * GAT_D) Wext[tid] = W[tid];
    if (tid < GAT_D)     Wext[2 * GAT_D + tid] = 0.0f;
    __syncthreads();

    const int base  = blockIdx.x * 128 + wave * 16;  // first node of this wave
    const int hiSel = lane >> 4;                     // 0: lanes 0-15, 1: lanes 16-31
    const int mrow  = lane & 15;
    const int bsel  = (mrow < 2 ? mrow : 2) * GAT_D; // W1 / W2 / zero row, once

    float* t = tile[wave];
    v8f c = {};

    for (int half = 0; half < 2; ++half) {
        const int h0 = half * 64;

#if HAVE_ASYNC_LDS
        if (half) {
            // WAR: async LDS writes must not pass the previous half's ds reads.
            asm volatile("s_wait_dscnt 0" ::: "memory");
        }
#endif
        // Stage 16 rows x 64 cols, coalesced: 32 lanes x 8B = one 256B row.
        #pragma unroll 4
        for (int r = 0; r < 16; ++r) {
            int grow = base + r;
            if (grow > N - 1) grow = N - 1;          // clamp tail (values unused)
            row_copy8(h + (size_t)grow * GAT_D + h0 + lane * 2,
                      &t[r * TILE_STRIDE + lane * 2]);
        }
#if HAVE_ASYNC_LDS
        __builtin_amdgcn_s_wait_asynccnt(0);
#endif

        // 16 WMMA steps of K=4. A layout (f32 16x4): lanes 0-15 hold K=kb,kb+1
        // for M=lane; lanes 16-31 hold K=kb+2,kb+3 for M=lane-16.
        // B layout (f32 4x16): VGPR v, lane L -> B[K=(L>=16?2:0)+v][Nc=L%16].
        #pragma unroll
        for (int kk = 0; kk < 16; ++kk) {
            const int kb = kk * 4 + hiSel * 2;
            v2f a, b;
            a.x = t[mrow * TILE_STRIDE + kb];
            a.y = t[mrow * TILE_STRIDE + kb + 1];
            b.x = Wext[bsel + h0 + kb];              // col 0=W1, col 1=W2, else 0
            b.y = Wext[bsel + h0 + kb + 1];
            // D = A*B + C  (neg_a, A, neg_b, B, c_mod, C, reuse_a, reuse_b)
            c = __builtin_amdgcn_wmma_f32_16x16x4_f32(
                    false, a, false, b, (short)0, c, false, false);
        }
    }

    // D layout (f32 16x16): VGPR r -> M=r (lanes 0-15) / M=8+r (lanes 16-31),
    // column Nc = lane%16.  Column 0 = s_src, column 1 = s_dst.
    if (mrow < 2) {
        float* outp  = (mrow == 0) ? ssrc : sdst;
        const int mb = base + hiSel * 8;
        #pragma unroll
        for (int r = 0; r < 8; ++r) {
            const int node = mb + r;
            if (node < N) outp[node] = c[r];
        }
    }
}

// ---------------------------------------------------------------------------
// Phase 2: sum of leaky_relu(s_src[g0]+s_dst[g1]+b) over edges with g0 == i.
// Streams g once (int2, coalesced); gathers hit L2 (s arrays are 400KB each).
// ---------------------------------------------------------------------------
__global__ __launch_bounds__(256) void gat_edge_sum_kernel(
    const int* __restrict__ g, const float* __restrict__ ssrc,
    const float* __restrict__ sdst, const float* __restrict__ bptr,
    const int* __restrict__ iptr, float* __restrict__ acc, int E)
{
    __shared__ float red[256];
    const float bb = bptr[0];
    const int  ii  = iptr[0];
    const int  stride = gridDim.x * blockDim.x;
    const int2* __restrict__ g2 = reinterpret_cast<const int2*>(g);

    float sum = 0.0f;
    for (int e = blockIdx.x * blockDim.x + threadIdx.x; e < E; e += stride) {
        __builtin_prefetch(&g2[e + stride], 0, 0);   // speculative global prefetch
        const int2 gg = g2[e];
        if (gg.x == ii) {
            const float x = ssrc[gg.x] + sdst[gg.y] + bb;
            sum += (x >= 0.0f) ? x : NEG_SLOPE * x;
        }
    }

    red[threadIdx.x] = sum;
    __syncthreads();
    #pragma unroll
    for (int s = 128; s > 0; s >>= 1) {
        if (threadIdx.x < s) red[threadIdx.x] += red[threadIdx.x + s];
        __syncthreads();
    }
    if (threadIdx.x == 0) atomicAdd(acc, red[0]);
}

// ---------------------------------------------------------------------------
// Phase 3: out = leaky_relu(s_src[i]+s_dst[j]+b) / sumneighbors
// ---------------------------------------------------------------------------
__global__ void gat_final_kernel(
    const float* __restrict__ ssrc, const float* __restrict__ sdst,
    const float* __restrict__ bptr, const int* __restrict__ iptr,
    const int* __restrict__ jptr, const float* __restrict__ acc,
    float* __restrict__ out)
{
    if (threadIdx.x == 0 && blockIdx.x == 0) {
        const float x = ssrc[iptr[0]] + sdst[jptr[0]] + bptr[0];
        const float e = (x >= 0.0f) ? x : NEG_SLOPE * x;
        out[0] = e / acc[0];
    }
}

extern "C" void kernel_launch(void* const* d_in, const int* in_sizes, int n_in,
                              void* d_out, int out_size, void* d_ws, size_t ws_size,
                              hipStream_t stream)
{
    const int*   g  = (const int*)  d_in[0];   // (E,2) int32
    const float* h  = (const float*)d_in[1];   // (N,128) f32
    const int*   ip = (const int*)  d_in[2];   // scalar i
    const int*   jp = (const int*)  d_in[3];   // scalar j
    const float* W  = (const float*)d_in[4];   // (1,256) f32
    const float* b  = (const float*)d_in[5];   // (1,) f32

    const int N = in_sizes[1] / GAT_D;
    const int E = in_sizes[0] / 2;

    float* ssrc = (float*)d_ws;                // N floats
    float* sdst = ssrc + N;                    // N floats
    float* acc  = sdst + N;                    // 1 float

    (void)hipMemsetAsync(acc, 0, sizeof(float), stream);   // graph-capture safe

    const int blocks1 = (N + 127) / 128;             // 128 nodes per block (8 waves x 16)
    gat_scores_kernel<<<blocks1, 256, 0, stream>>>(h, W, ssrc, sdst, N);

    const int blocks2 = 1024;                        // grid-stride over 3.2M edges
    gat_edge_sum_kernel<<<blocks2, 256, 0, stream>>>(g, ssrc, sdst, b, ip, acc, E);

    gat_final_kernel<<<1, 32, 0, stream>>>(ssrc, sdst, b, ip, jp, acc, (float*)d_out);
}